// Embedding4bit_26972394619030
// MI455X (gfx1250) — compile-verified
//
#include <hip/hip_runtime.h>

// NF4 embedding dequant for MI455X (gfx1250, wave32).
// Memory-bound gather: no matmul -> WMMA not applicable. Uses CDNA5 async
// global->LDS copy engine (ASYNCcnt) for row staging + NT b128 stores.
// Absmax is wave-uniform per row -> SMEM scalar loads + v_cndmask select
// (no ds_bpermute shuffles on the DS pipe).

typedef __attribute__((ext_vector_type(4))) float float4v;

#define EMBED_DIM       2048
#define ROW_BYTES       (EMBED_DIM / 2)   // 1024 packed bytes per row
#define NUM_BLOCKS      32                // absmax blocks per row (blocksize 64)
#define WAVES_PER_BLOCK 8
#define BLOCK_THREADS   256

__constant__ float NF4_VALS[16] = {
    -1.0f, -0.6961928009986877f, -0.5250730514526367f, -0.39491748809814453f,
    -0.28444138169288635f, -0.18477343022823334f, -0.09105003625154495f, 0.0f,
    0.07958029955625534f, 0.16093020141124725f, 0.24611230194568634f,
    0.33791524171829224f, 0.44070982933044434f, 0.5626170039176941f,
    0.7229568362236023f, 1.0f};

__global__ __launch_bounds__(BLOCK_THREADS) void nf4_embed_kernel(
    const int* __restrict__ ids,           // [nrows] token ids
    const unsigned char* __restrict__ wp,  // [VOCAB, 1024] packed nibbles
    const float* __restrict__ am,          // [VOCAB, 32] absmax
    float* __restrict__ out,               // [nrows, 2048] (fp16-rounded values)
    int nrows)
{
    // 256-entry byte-pair LUT: lut[b] = {nf4[b>>4], nf4[b&15]}  (2 KB)
    __shared__ float2 lut[256];
    // Per-wave double-buffered staging for one packed row (16 KB total)
    __shared__ alignas(16) unsigned char stage[WAVES_PER_BLOCK][2][ROW_BYTES];

    const int tid = threadIdx.x;
    if (tid < 256) {
        lut[tid] = make_float2(NF4_VALS[tid >> 4], NF4_VALS[tid & 15]);
    }
    __syncthreads();

    const int lane = tid & 31;
    const int wv   = tid >> 5;
    const int gw   = blockIdx.x * WAVES_PER_BLOCK + wv;
    const int nW   = gridDim.x * WAVES_PER_BLOCK;
    const bool loHalf = (lane < 16);       // selects even/odd absmax block

    // Low 32 bits of the generic address of an LDS object == LDS byte offset
    // (generic LDS addr = {SHARED_BASE, offset[31:0]} per CDNA5 ISA 10.2).
    const unsigned int stageOff =
        (unsigned int)(uintptr_t)(&stage[wv][0][0]);

    // Async row prefetch: 32 lanes x 16 B x 2 issues = 1024 B -> LDS buffer b.
    // INST_OFFSET applies to both the global and LDS address (ISA 10, async).
    auto prefetch = [&](int row, int b) {
        const int tok = ids[row];
        const unsigned long long ga =
            (unsigned long long)(uintptr_t)(wp + (size_t)tok * ROW_BYTES + lane * 16);
        const unsigned int la = stageOff + (unsigned int)b * ROW_BYTES + lane * 16;
        asm volatile("global_load_async_to_lds_b128 %0, %1, off"
                     :: "v"(la), "v"(ga) : "memory");
        asm volatile("global_load_async_to_lds_b128 %0, %1, off offset:512"
                     :: "v"(la), "v"(ga) : "memory");
    };

    int row = gw;
    int buf = 0;
    if (row < nrows) prefetch(row, 0);

    for (; row < nrows; row += nW, buf ^= 1) {
        const int nxt = row + nW;
        const bool hasNext = (nxt < nrows);
        if (hasNext) prefetch(nxt, buf ^ 1);   // overlap fetch with decode

        const int tok = ids[row];              // uniform -> scalar load
        // Whole absmax row is wave-uniform: SMEM scalar loads (KMcnt path),
        // no vector-memory gather and no cross-lane shuffle needed.
        const float* amrow = am + (size_t)tok * NUM_BLOCKS;

        // ASYNCcnt completes in order: with next row's 2 issues outstanding,
        // waiting to <=2 guarantees the current buffer is fully written.
        if (hasNext) {
            asm volatile("s_wait_asynccnt 2" ::: "memory");
        } else {
            asm volatile("s_wait_asynccnt 0" ::: "memory");
        }

        const unsigned short* sp = (const unsigned short*)&stage[wv][buf][0];
        float* orow = out + (size_t)row * EMBED_DIM;

        // 16 iterations x (32 lanes x 2 packed bytes) covers the 1024-byte row.
        // Lane l, iter i: bytes p = i*64 + 2l -> output elems i*128 + 4l .. +3,
        // absmax block = 2i + (l>>4); store is unit-stride 512 B / instruction.
        #pragma unroll
        for (int i = 0; i < 16; ++i) {
            const unsigned int w = sp[i * 32 + lane];        // 2 packed bytes
            const float a0 = amrow[2 * i];                   // uniform s_load
            const float a1 = amrow[2 * i + 1];               // uniform s_load
            const float s = loHalf ? a0 : a1;                // v_cndmask
            const float2 q0 = lut[w & 255];                  // elems +0,+1
            const float2 q1 = lut[(w >> 8) & 255];           // elems +2,+3
            float4v r;
            r.x = (float)(_Float16)(q0.x * s);
            r.y = (float)(_Float16)(q0.y * s);
            r.z = (float)(_Float16)(q1.x * s);
            r.w = (float)(_Float16)(q1.y * s);
            __builtin_nontemporal_store(r, (float4v*)(orow + i * 128 + lane * 4));
        }
    }
}

extern "C" void kernel_launch(void* const* d_in, const int* in_sizes, int n_in,
                              void* d_out, int out_size, void* d_ws, size_t ws_size,
                              hipStream_t stream) {
    const int* ids           = (const int*)d_in[0];           // [4*8192] int32
    const unsigned char* wp  = (const unsigned char*)d_in[1]; // [128000*1024] u8
    const float* am          = (const float*)d_in[2];         // [128000*32] f32
    float* out               = (float*)d_out;                 // fp16-rounded f32

    const int nrows  = in_sizes[0];                           // 32768
    const int blocks = (nrows + WAVES_PER_BLOCK - 1) / WAVES_PER_BLOCK;
    nf4_embed_kernel<<<blocks, BLOCK_THREADS, 0, stream>>>(ids, wp, am, out, nrows);
}